// CrossHyperConvBlock_43344809951392
// MI455X (gfx1250) — compile-verified
//
#include <hip/hip_runtime.h>

typedef float v2f __attribute__((ext_vector_type(2)));
typedef float v8f __attribute__((ext_vector_type(8)));

namespace {
constexpr int Dd = 1024;   // feature dim
constexpr int Bb = 32;     // batch
constexpr int Tt = 1024;   // time
constexpr int NK = 3072;   // d * K
constexpr int TCHUNK = 4;  // timesteps per block in fused kernel
constexpr float LN_EPS = 1e-5f;
}

// ---------------------------------------------------------------------------
// W = cond(32x1024) @ gen_w.T(1024x3072) + gen_b, full-precision WMMA f32 K=4.
// One wave computes one 16x16 tile; 8 waves per block cover 8 N-tiles.
// A 16x4 f32 layout: lanes 0-15 -> M=lane, {K=k,k+1}; lanes 16-31 -> {K=k+2,k+3}
// B 4x16 f32 layout mirrors A with N in place of M. B(k,n) = gen_w[n,k].
// C/D 16x16 f32: VGPR r, lanes 0-15 -> M=r, N=lane; lanes 16-31 -> M=r+8.
// ---------------------------------------------------------------------------
__global__ void __launch_bounds__(256)
hypergen_wmma_kernel(const float* __restrict__ cond,   // (32,1024)
                     const float* __restrict__ genw,   // (3072,1024) row-major
                     const float* __restrict__ genb,   // (3072)
                     float* __restrict__ Wout)         // (32,3072)
{
    const int wave = threadIdx.x >> 5;
    const int lane = threadIdx.x & 31;
    const int l    = lane & 15;
    const int half = lane >> 4;                 // 0: K even pair, 1: K odd pair
    const int n0   = (blockIdx.x * 8 + wave) * 16;
    const int m0   = blockIdx.y * 16;

    const float* __restrict__ arow = cond + (size_t)(m0 + l) * Dd;
    const float* __restrict__ brow = genw + (size_t)(n0 + l) * Dd;

    v8f acc = {};
#pragma unroll 4
    for (int k = 0; k < Dd; k += 4) {
        const int ka = k + half * 2;
        v2f a, b;
        a.x = arow[ka];
        a.y = arow[ka + 1];
        b.x = brow[ka];
        b.y = brow[ka + 1];
        acc = __builtin_amdgcn_wmma_f32_16x16x4_f32(false, a, false, b,
                                                    (short)0, acc, false, false);
    }

    const int col    = n0 + l;
    const float bias = genb[col];
#pragma unroll
    for (int r = 0; r < 8; ++r) {
        const int row = m0 + r + half * 8;
        Wout[(size_t)row * NK + col] = acc[r] + bias;
    }
}

// ---------------------------------------------------------------------------
// Fused: depthwise 3-tap conv (both streams), relu(alpha) 2x2 mix, residual,
// and layernorm over d for C and R simultaneously. One (4-timestep, b) slab
// per 256-thread block; rolling register window over t to reuse rows.
// ---------------------------------------------------------------------------
__global__ void __launch_bounds__(256)
fused_conv_mix_ln_kernel(const float* __restrict__ C, const float* __restrict__ R,
                         const float* __restrict__ Wgc, const float* __restrict__ Wcg,
                         const float* __restrict__ alpha,
                         const float* __restrict__ g_c, const float* __restrict__ b_c,
                         const float* __restrict__ g_r, const float* __restrict__ b_r,
                         float* __restrict__ Cout, float* __restrict__ Rout)
{
    const int b    = blockIdx.y;
    const int t0   = blockIdx.x * TCHUNK;
    const int c    = threadIdx.x * 4;
    const int lane = threadIdx.x & 31;
    const int wave = threadIdx.x >> 5;
    const size_t strideT = (size_t)Bb * Dd;
    const size_t rowBase = (size_t)b * Dd + c;

    // conv taps: W[b, c+i, j] = Wflat[b*3072 + (c+i)*3 + j]; 12 floats, 16B aligned
    float wg[12], wc[12];
    {
        const size_t wb = (size_t)b * NK + (size_t)c * 3;
        *(float4*)&wg[0] = *(const float4*)(Wgc + wb);
        *(float4*)&wg[4] = *(const float4*)(Wgc + wb + 4);
        *(float4*)&wg[8] = *(const float4*)(Wgc + wb + 8);
        *(float4*)&wc[0] = *(const float4*)(Wcg + wb);
        *(float4*)&wc[4] = *(const float4*)(Wcg + wb + 4);
        *(float4*)&wc[8] = *(const float4*)(Wcg + wb + 8);
    }
    const float a00 = fmaxf(alpha[0], 0.f), a01 = fmaxf(alpha[1], 0.f);
    const float a10 = fmaxf(alpha[2], 0.f), a11 = fmaxf(alpha[3], 0.f);

    float gc[4], bc[4], gr[4], br[4];
    *(float4*)gc = *(const float4*)(g_c + c);
    *(float4*)bc = *(const float4*)(b_c + c);
    *(float4*)gr = *(const float4*)(g_r + c);
    *(float4*)br = *(const float4*)(b_r + c);

    float cprev[4] = {0,0,0,0}, rprev[4] = {0,0,0,0};
    float ccur[4],  rcur[4];
    if (t0 > 0) {
        *(float4*)cprev = *(const float4*)(C + (size_t)(t0 - 1) * strideT + rowBase);
        *(float4*)rprev = *(const float4*)(R + (size_t)(t0 - 1) * strideT + rowBase);
    }
    *(float4*)ccur = *(const float4*)(C + (size_t)t0 * strideT + rowBase);
    *(float4*)rcur = *(const float4*)(R + (size_t)t0 * strideT + rowBase);

    __shared__ float red[8][4];

    for (int tt = 0; tt < TCHUNK; ++tt) {
        const int t = t0 + tt;
        float cnext[4] = {0,0,0,0}, rnext[4] = {0,0,0,0};
        if (t < Tt - 1) {
            *(float4*)cnext = *(const float4*)(C + (size_t)(t + 1) * strideT + rowBase);
            *(float4*)rnext = *(const float4*)(R + (size_t)(t + 1) * strideT + rowBase);
        }

        float cmix[4], rmix[4];
        float s0 = 0.f, s1 = 0.f, s2 = 0.f, s3 = 0.f;
#pragma unroll
        for (int i = 0; i < 4; ++i) {
            // Y[t] = W[0]*X[t-1] + W[1]*X[t] + W[2]*X[t+1]  (cross-correlation)
            const float ch = wg[3*i] * cprev[i] + wg[3*i+1] * ccur[i] + wg[3*i+2] * cnext[i];
            const float rh = wc[3*i] * rprev[i] + wc[3*i+1] * rcur[i] + wc[3*i+2] * rnext[i];
            const float cm = a00 * ch + a01 * rh + ccur[i];   // mix + residual
            const float rm = a10 * ch + a11 * rh + rcur[i];
            cmix[i] = cm; rmix[i] = rm;
            s0 += cm; s1 += cm * cm; s2 += rm; s3 += rm * rm;
        }

        // wave32 butterfly reduction
#pragma unroll
        for (int off = 16; off > 0; off >>= 1) {
            s0 += __shfl_xor(s0, off, 32);
            s1 += __shfl_xor(s1, off, 32);
            s2 += __shfl_xor(s2, off, 32);
            s3 += __shfl_xor(s3, off, 32);
        }
        __syncthreads();   // protect red[][] from previous iteration's readers
        if (lane == 0) {
            red[wave][0] = s0; red[wave][1] = s1; red[wave][2] = s2; red[wave][3] = s3;
        }
        __syncthreads();
        float T0 = 0.f, T1 = 0.f, T2 = 0.f, T3 = 0.f;
#pragma unroll
        for (int w = 0; w < 8; ++w) {
            T0 += red[w][0]; T1 += red[w][1]; T2 += red[w][2]; T3 += red[w][3];
        }
        const float invD = 1.0f / (float)Dd;
        const float muc = T0 * invD;
        const float mur = T2 * invD;
        const float ivc = rsqrtf(T1 * invD - muc * muc + LN_EPS);
        const float ivr = rsqrtf(T3 * invD - mur * mur + LN_EPS);

        float co[4], ro[4];
#pragma unroll
        for (int i = 0; i < 4; ++i) {
            co[i] = (cmix[i] - muc) * ivc * gc[i] + bc[i];
            ro[i] = (rmix[i] - mur) * ivr * gr[i] + br[i];
        }
        const size_t ob = (size_t)t * strideT + rowBase;
        *(float4*)(Cout + ob) = *(float4*)co;
        *(float4*)(Rout + ob) = *(float4*)ro;

#pragma unroll
        for (int i = 0; i < 4; ++i) {
            cprev[i] = ccur[i]; ccur[i] = cnext[i];
            rprev[i] = rcur[i]; rcur[i] = rnext[i];
        }
    }
}

// ---------------------------------------------------------------------------
extern "C" void kernel_launch(void* const* d_in, const int* in_sizes, int n_in,
                              void* d_out, int out_size, void* d_ws, size_t ws_size,
                              hipStream_t stream) {
    const float* C     = (const float*)d_in[0];
    const float* R     = (const float*)d_in[1];
    const float* ggw   = (const float*)d_in[2];   // gen_gc_w (3072,1024)
    const float* ggb   = (const float*)d_in[3];   // gen_gc_b (3072)
    const float* gcw   = (const float*)d_in[4];   // gen_cg_w
    const float* gcb   = (const float*)d_in[5];   // gen_cg_b
    const float* alpha = (const float*)d_in[6];   // (2,2)
    const float* lncg  = (const float*)d_in[7];
    const float* lncb  = (const float*)d_in[8];
    const float* lnrg  = (const float*)d_in[9];
    const float* lnrb  = (const float*)d_in[10];

    float* out  = (float*)d_out;
    float* Cout = out;
    float* Rout = out + (size_t)Tt * Bb * Dd;

    float* Wgc = (float*)d_ws;                    // (32,3072)
    float* Wcg = Wgc + (size_t)Bb * NK;           // (32,3072)

    const float* cond_gc = R + (size_t)(Tt - 1) * Bb * Dd;  // R[-1] -> (B,d)
    const float* cond_cg = C + (size_t)(Tt - 1) * Bb * Dd;  // C[-1] -> (B,d)

    dim3 gemmGrid(NK / (16 * 8), Bb / 16);        // (24, 2), 8 waves/block
    hypergen_wmma_kernel<<<gemmGrid, 256, 0, stream>>>(cond_gc, ggw, ggb, Wgc);
    hypergen_wmma_kernel<<<gemmGrid, 256, 0, stream>>>(cond_cg, gcw, gcb, Wcg);

    dim3 fusedGrid(Tt / TCHUNK, Bb);              // (256, 32)
    fused_conv_mix_ln_kernel<<<fusedGrid, 256, 0, stream>>>(
        C, R, Wgc, Wcg, alpha, lncg, lncb, lnrg, lnrb, Cout, Rout);
}